// SimpleQMixer_28750511080090
// MI455X (gfx1250) — compile-verified
//
#include <hip/hip_runtime.h>
#include <hip/hip_bf16.h>

#define BATCH 4096
#define SDIM 4096
#define EMBED 64
#define NAGENTS 8
#define NCOLS 256                      // 4 fused weight chunks of 64 cols
#define SROW_STRIDE (8 * 4 * 32 * 32)  // states row stride in floats (A=8,4,32,32)
#define KC 64                          // K-chunk staged in LDS

typedef float v2f __attribute__((ext_vector_type(2)));
typedef float v8f __attribute__((ext_vector_type(8)));
typedef int   v4i __attribute__((ext_vector_type(4)));

#define GA_AS  __attribute__((address_space(1)))
#define LDS_AS __attribute__((address_space(3)))

#if defined(__has_builtin)
#if __has_builtin(__builtin_amdgcn_global_load_async_to_lds_b128)
#define HAVE_ASYNC_LDS 1
#endif
#if __has_builtin(__builtin_amdgcn_s_wait_asynccnt)
#define HAVE_WAIT_ASYNC_BUILTIN 1
#endif
#endif

// One b128 slice of the cooperative A-chunk copy (global -> LDS).
// Builtin signature (from hipcc diagnostic): (v4i addrspace(1)*, v4i addrspace(3)*, imm, imm)
__device__ __forceinline__ void qmix_copy_a(const float* __restrict__ g,
                                            float* __restrict__ l) {
#if defined(HAVE_ASYNC_LDS)
    __builtin_amdgcn_global_load_async_to_lds_b128(
        (GA_AS v4i*)g, (LDS_AS v4i*)l, 0, 0);
#else
    *(float4*)l = *(const float4*)g;   // synchronous fallback
#endif
}

__device__ __forceinline__ void qmix_wait_async() {
#if defined(HAVE_ASYNC_LDS)
#if defined(HAVE_WAIT_ASYNC_BUILTIN)
    __builtin_amdgcn_s_wait_asynccnt(0);
#else
    asm volatile("s_wait_asynccnt 0" ::: "memory");
#endif
#endif
}

// ---------------------------------------------------------------------------
// Weight packing: rewrite the four (SDIM x 64) f32 weights into WMMA-fragment
// order so each lane's B fragment (rows kk, kk+1 at one column) is one
// contiguous float2 -> a single global_load_b64.
// Packed index: [((k>>2)*2 + hi) * NCOLS + gcol] with hi = lane>>4.
// ---------------------------------------------------------------------------
__global__ __launch_bounds__(256)
void qmix_pack_w(const float* __restrict__ w0, const float* __restrict__ w1,
                 const float* __restrict__ w2, const float* __restrict__ w3,
                 v2f* __restrict__ Wp) {
    const int idx  = blockIdx.x * 256 + threadIdx.x;
    const int gcol = idx & (NCOLS - 1);
    const int rest = idx >> 8;
    const int hi   = rest & 1;
    const int kp   = rest >> 1;
    const int chunk = gcol >> 6;
    const int col   = gcol & 63;
    const float* __restrict__ W;
    switch (chunk) {
        case 0:  W = w0; break;
        case 1:  W = w1; break;
        case 2:  W = w2; break;
        default: W = w3; break;
    }
    const int kk = 4 * kp + 2 * hi;
    v2f v;
    v.x = W[(size_t)kk * EMBED + col];
    v.y = W[(size_t)(kk + 1) * EMBED + col];
    Wp[idx] = v;
}

// ---------------------------------------------------------------------------
// Stage 1: fused hypernetwork first layers, packed B + LDS-staged A.
// Block = 16 batch rows x 256 cols; 8 waves; wave = two 16x16 WMMA tiles.
// A chunk (16 x KC f32 = 4KB) double-buffered in LDS, filled with async
// global->LDS b128 copies overlapped with WMMA compute on the other buffer.
// ---------------------------------------------------------------------------
__global__ __launch_bounds__(256)
void qmix_hyper_gemm_lds(const float* __restrict__ states,
                         const v2f* __restrict__ Wp,
                         const float* __restrict__ b0p, const float* __restrict__ b1p,
                         const float* __restrict__ b2p, const float* __restrict__ b3p,
                         float* __restrict__ H) {
    __shared__ float As[2][16 * KC];   // 2 x 4KB double buffer

    const int tid   = threadIdx.x;
    const int wave  = tid >> 5;
    const int lane  = tid & 31;
    const int m0    = blockIdx.x << 4;
    const int chunk = wave >> 1;
    const int cbase = (wave & 1) * 32;

    const float* __restrict__ Bv;
    switch (chunk) {
        case 0:  Bv = b0p; break;
        case 1:  Bv = b1p; break;
        case 2:  Bv = b2p; break;
        default: Bv = b3p; break;
    }

    const int m  = lane & 15;
    const int hi = lane >> 4;
    const int n  = lane & 15;
    const int col0  = cbase + n;
    const int col1  = cbase + 16 + n;
    const int gcol0 = chunk * EMBED + col0;
    const int gcol1 = chunk * EMBED + col1;

    const v2f* __restrict__ pb0 = Wp + (size_t)hi * NCOLS + gcol0;
    const v2f* __restrict__ pb1 = Wp + (size_t)hi * NCOLS + gcol1;

    // Cooperative A loader: thread t copies 4 floats of row (t>>4).
    const int lrow = tid >> 4;
    const int lxi  = (tid & 15) * 4;
    const float* __restrict__ gA =
        states + (size_t)(m0 + lrow) * SROW_STRIDE + lxi;
    float* __restrict__ lA = &As[0][0] + lrow * KC + lxi;   // buffer 0 slot

    v8f c0 = {0.f, 0.f, 0.f, 0.f, 0.f, 0.f, 0.f, 0.f};
    v8f c1 = {0.f, 0.f, 0.f, 0.f, 0.f, 0.f, 0.f, 0.f};

    const int nchunk = SDIM / KC;      // 64
    qmix_copy_a(gA, lA);               // prologue: chunk 0 -> buf 0

    for (int c = 0; c < nchunk; ++c) {
        qmix_wait_async();
        __syncthreads();               // buf[c&1] ready for all waves

        if (c + 1 < nchunk)            // overlap next chunk with compute
            qmix_copy_a(gA + (size_t)(c + 1) * KC,
                        lA + ((c + 1) & 1) * (16 * KC));

        const float* __restrict__ al = &As[c & 1][m * KC + 2 * hi];
        const int kbase = c * KC;
#pragma unroll 8
        for (int kl = 0; kl < KC; kl += 4) {
            v2f a  = *(const v2f*)(al + kl);                       // ds_load_b64
            v2f b0 = pb0[(size_t)((kbase + kl) >> 2) * (2 * NCOLS)];
            v2f b1 = pb1[(size_t)((kbase + kl) >> 2) * (2 * NCOLS)];
            c0 = __builtin_amdgcn_wmma_f32_16x16x4_f32(false, a, false, b0,
                                                       (short)0, c0, false, false);
            c1 = __builtin_amdgcn_wmma_f32_16x16x4_f32(false, a, false, b1,
                                                       (short)0, c1, false, false);
        }
    }

    const bool do_relu = (chunk != 1);
    const float bias0 = Bv[col0];
    const float bias1 = Bv[col1];
#pragma unroll
    for (int i = 0; i < 8; ++i) {
        const int row = m0 + i + 8 * hi;
        float v0 = c0[i] + bias0;
        float v1 = c1[i] + bias1;
        if (do_relu) { v0 = fmaxf(v0, 0.f); v1 = fmaxf(v1, 0.f); }
        H[(size_t)row * NCOLS + gcol0] = v0;
        H[(size_t)row * NCOLS + gcol1] = v1;
    }
}

// ---------------------------------------------------------------------------
// Stage 1 fallback (direct strided B loads) for small ws_size.
// ---------------------------------------------------------------------------
__global__ __launch_bounds__(256)
void qmix_hyper_gemm_direct(const float* __restrict__ states,
                            const float* __restrict__ w0, const float* __restrict__ b0p,
                            const float* __restrict__ w1, const float* __restrict__ b1p,
                            const float* __restrict__ w2, const float* __restrict__ b2p,
                            const float* __restrict__ w3, const float* __restrict__ b3p,
                            float* __restrict__ H) {
    const int wave  = threadIdx.x >> 5;
    const int lane  = threadIdx.x & 31;
    const int m0    = blockIdx.x << 4;
    const int chunk = wave >> 1;
    const int cbase = (wave & 1) * 32;

    const float* __restrict__ W;
    const float* __restrict__ Bv;
    switch (chunk) {
        case 0:  W = w0; Bv = b0p; break;
        case 1:  W = w1; Bv = b1p; break;
        case 2:  W = w2; Bv = b2p; break;
        default: W = w3; Bv = b3p; break;
    }

    const int m  = lane & 15;
    const int hi = lane >> 4;
    const int n  = lane & 15;
    const float* __restrict__ arow = states + (size_t)(m0 + m) * SROW_STRIDE;
    const int col0 = cbase + n;
    const int col1 = cbase + 16 + n;

    v8f c0 = {0.f, 0.f, 0.f, 0.f, 0.f, 0.f, 0.f, 0.f};
    v8f c1 = {0.f, 0.f, 0.f, 0.f, 0.f, 0.f, 0.f, 0.f};

#pragma unroll 4
    for (int k = 0; k < SDIM; k += 4) {
        const int kk = k + 2 * hi;
        v2f a = *(const v2f*)(arow + kk);
        v2f bb0, bb1;
        bb0.x = W[(size_t)kk       * EMBED + col0];
        bb0.y = W[(size_t)(kk + 1) * EMBED + col0];
        bb1.x = W[(size_t)kk       * EMBED + col1];
        bb1.y = W[(size_t)(kk + 1) * EMBED + col1];
        c0 = __builtin_amdgcn_wmma_f32_16x16x4_f32(false, a, false, bb0,
                                                   (short)0, c0, false, false);
        c1 = __builtin_amdgcn_wmma_f32_16x16x4_f32(false, a, false, bb1,
                                                   (short)0, c1, false, false);
    }

    const bool do_relu = (chunk != 1);
    const float bias0 = Bv[col0];
    const float bias1 = Bv[col1];
    const int gcol0 = chunk * EMBED + col0;
    const int gcol1 = chunk * EMBED + col1;
#pragma unroll
    for (int i = 0; i < 8; ++i) {
        const int row = m0 + i + 8 * hi;
        float v0 = c0[i] + bias0;
        float v1 = c1[i] + bias1;
        if (do_relu) { v0 = fmaxf(v0, 0.f); v1 = fmaxf(v1, 0.f); }
        H[(size_t)row * NCOLS + gcol0] = v0;
        H[(size_t)row * NCOLS + gcol1] = v1;
    }
}

// ---------------------------------------------------------------------------
// Stage 2: second hypernetwork layers + mixing, one 64-thread block per row.
// ---------------------------------------------------------------------------
__global__ __launch_bounds__(64)
void qmix_mix(const float* __restrict__ H,
              const float* __restrict__ agent_qs,
              const float* __restrict__ hw1_w2, const float* __restrict__ hw1_b2,
              const float* __restrict__ hw2_w2, const float* __restrict__ hw2_b2,
              const float* __restrict__ hb2_w2, const float* __restrict__ hb2_b2,
              float* __restrict__ out) {
    __shared__ float sh[NCOLS];
    __shared__ float red[2 * EMBED];
    const int b = blockIdx.x;
    const int e = threadIdx.x;

#pragma unroll
    for (int i = 0; i < 4; ++i)
        sh[e + 64 * i] = H[(size_t)b * NCOLS + e + 64 * i];
    __syncthreads();

    float q[NAGENTS];
#pragma unroll
    for (int a = 0; a < NAGENTS; ++a) q[a] = agent_qs[b * NAGENTS + a];

    float w1mix = 0.f;
#pragma unroll
    for (int a = 0; a < NAGENTS; ++a) {
        float acc = hw1_b2[a * EMBED + e];
#pragma unroll 8
        for (int k = 0; k < EMBED; ++k)
            acc = fmaf(sh[k], hw1_w2[k * (NAGENTS * EMBED) + a * EMBED + e], acc);
        w1mix = fmaf(q[a], fabsf(acc), w1mix);
    }

    float acc2 = hw2_b2[e];
#pragma unroll 8
    for (int k = 0; k < EMBED; ++k)
        acc2 = fmaf(sh[128 + k], hw2_w2[k * EMBED + e], acc2);
    const float w2v = fabsf(acc2);

    const float x = w1mix + sh[64 + e];
    const float hidden = (x > 0.f) ? x : (expf(x) - 1.f);

    red[e]      = hidden * w2v;
    red[64 + e] = sh[192 + e] * hb2_w2[e];
    __syncthreads();
    if (e == 0) {
        float s = hb2_b2[0];
        for (int i = 0; i < 2 * EMBED; ++i) s += red[i];
        out[b] = s;
    }
}

extern "C" void kernel_launch(void* const* d_in, const int* in_sizes, int n_in,
                              void* d_out, int out_size, void* d_ws, size_t ws_size,
                              hipStream_t stream) {
    const float* agent_qs = (const float*)d_in[0];
    const float* states   = (const float*)d_in[1];
    const float* hw1_w1   = (const float*)d_in[2];
    const float* hw1_b1   = (const float*)d_in[3];
    const float* hw1_w2   = (const float*)d_in[4];
    const float* hw1_b2   = (const float*)d_in[5];
    const float* hb1_w    = (const float*)d_in[6];
    const float* hb1_b    = (const float*)d_in[7];
    const float* hw2_w1   = (const float*)d_in[8];
    const float* hw2_b1   = (const float*)d_in[9];
    const float* hw2_w2   = (const float*)d_in[10];
    const float* hw2_b2   = (const float*)d_in[11];
    const float* hb2_w1   = (const float*)d_in[12];
    const float* hb2_b1   = (const float*)d_in[13];
    const float* hb2_w2   = (const float*)d_in[14];
    const float* hb2_b2   = (const float*)d_in[15];

    float* H   = (float*)d_ws;
    float* out = (float*)d_out;
    const size_t h_elems  = (size_t)BATCH * NCOLS;
    const size_t wp_elems = (size_t)SDIM * NCOLS;
    const bool use_packed = ws_size >= (h_elems + wp_elems) * sizeof(float);

    if (use_packed) {
        v2f* Wp = (v2f*)((float*)d_ws + h_elems);
        qmix_pack_w<<<(int)(wp_elems / 2 / 256), 256, 0, stream>>>(
            hw1_w1, hb1_w, hw2_w1, hb2_w1, Wp);
        qmix_hyper_gemm_lds<<<BATCH / 16, 256, 0, stream>>>(
            states, Wp, hw1_b1, hb1_b, hw2_b1, hb2_b1, H);
    } else {
        qmix_hyper_gemm_direct<<<BATCH / 16, 256, 0, stream>>>(
            states,
            hw1_w1, hw1_b1, hb1_w, hb1_b,
            hw2_w1, hw2_b1, hb2_w1, hb2_b1,
            H);
    }

    qmix_mix<<<BATCH, 64, 0, stream>>>(
        H, agent_qs,
        hw1_w2, hw1_b2,
        hw2_w2, hw2_b2,
        hb2_w2, hb2_b2,
        out);
}